// Evaluation_78383153152424
// MI455X (gfx1250) — compile-verified
//
#include <hip/hip_runtime.h>
#include <hip/hip_bf16.h>

typedef __attribute__((ext_vector_type(2))) float v2f;
typedef __attribute__((ext_vector_type(8))) float v8f;

// Problem dimensions (fixed by the reference)
#define NB 2
#define NG 8
#define ND 32
#define NH 128
#define NW 160
#define DHW (ND * NH * NW)          // 655360
#define NPTS (NB * DHW)             // 1310720
#define TILES_PER_WAVE 4            // 16 points per tile -> 64 points per wave

// ds_swizzle_b32 group-of-32 SWAPX16: xor_mask=0x10, or_mask=0, and_mask=0x1f
#define SWIZ_SWAPX16 0x401F

__device__ __forceinline__ float half_swap(float v) {
    return __int_as_float(
        __builtin_amdgcn_ds_swizzle(__float_as_int(v), SWIZ_SWAPX16));
}

// ---------------------------------------------------------------------------
// Kernel 1: per-point MLP (8 -> 16 -> 8 -> 1) via V_WMMA_F32_16X16X4_F32.
// One wave handles 16 points per tile:
//   layer1: D16x16 = w0(16x8) * X(8x16)    -> 2 WMMAs (K split 4+4)
//   layer2: D16x16 = w1p(16x16) * H(16x16) -> 4 WMMAs (K split 4*4)
//   layer3: 1x8 dot on lanes 0..15 (rows 0..7 of C/D live there)
// ---------------------------------------------------------------------------
__global__ __launch_bounds__(256) void mlp_sim_kernel(
    const float* __restrict__ x1,
    const float* __restrict__ w0,
    const float* __restrict__ bn0_scale, const float* __restrict__ bn0_bias,
    const float* __restrict__ w1,
    const float* __restrict__ bn1_scale, const float* __restrict__ bn1_bias,
    const float* __restrict__ w_sim, const float* __restrict__ b_sim,
    float* __restrict__ sim)
{
    const int lane = threadIdx.x & 31;
    const int wib  = threadIdx.x >> 5;                 // wave in block (0..7)
    const int gwave = blockIdx.x * 8 + wib;
    const int hi   = lane >> 4;                        // 0: lanes 0-15, 1: lanes 16-31
    const int col  = lane & 15;                        // M (for A) / N (for B,C,D)

    // ---- A operand, layer 1: w0 is (16 x 8) row-major.
    // A 16x4 layout: lanes 0-15 {K0,K1}, lanes 16-31 {K2,K3}, M = lane&15.
    const int kb = hi * 2;
    v2f a0, a1;
    a0.x = w0[col * 8 + kb + 0];
    a0.y = w0[col * 8 + kb + 1];
    a1.x = w0[col * 8 + 4 + kb + 0];
    a1.y = w0[col * 8 + 4 + kb + 1];

    // ---- A operand, layer 2: w1 is (8 x 16) row-major, zero-padded to 16 rows.
    v2f aw[4];
#pragma unroll
    for (int j = 0; j < 4; ++j) {
        const int k0 = j * 4 + kb;
        aw[j].x = (col < 8) ? w1[col * 16 + k0 + 0] : 0.0f;
        aw[j].y = (col < 8) ? w1[col * 16 + k0 + 1] : 0.0f;
    }

    // ---- BN params per C/D row: row M = v + 8*hi for VGPR v.
    float s0[8], bi0[8], s1[8], bi1[8], ws[8];
    const int rb = 8 * hi;
#pragma unroll
    for (int v = 0; v < 8; ++v) {
        s0[v]  = bn0_scale[rb + v];
        bi0[v] = bn0_bias[rb + v];
        const int r = (rb + v) & 7;    // rows 8..15 of layer-2 output are padding
        s1[v]  = bn1_scale[r];
        bi1[v] = bn1_bias[r];
        ws[v]  = w_sim[v];
    }
    const float bs = b_sim[0];

#pragma unroll
    for (int t = 0; t < TILES_PER_WAVE; ++t) {
        const int base = (gwave * TILES_PER_WAVE + t) * 16;   // first of 16 points
        const int p    = base + col;                          // this column's point
        const int b    = p / DHW;
        const int rem  = p - b * DHW;                         // (d,h,w) flat

        // ---- B operand, layer 1: X chunk (4 channels x 16 points).
        // B 4x16 layout mirrors A: lanes 0-15 {K0,K1}, lanes 16-31 {K2,K3}.
        const float* xp = x1 + (size_t)(b * NG + kb) * DHW + rem;
        v2f b0, b1;
        b0.x = xp[0];
        b0.y = xp[DHW];
        b1.x = xp[4 * (size_t)DHW];
        b1.y = xp[5 * (size_t)DHW];
        __builtin_prefetch(xp + 16, 0, 3);   // next tile, same channels (WGP scope)

        // ---- layer 1 GEMM + BN + ReLU
        v8f c = {};
        c = __builtin_amdgcn_wmma_f32_16x16x4_f32(false, a0, false, b0,
                                                  (short)0, c, false, false);
        c = __builtin_amdgcn_wmma_f32_16x16x4_f32(false, a1, false, b1,
                                                  (short)0, c, false, false);
        float h[8];
#pragma unroll
        for (int v = 0; v < 8; ++v)
            h[v] = fmaxf(fmaf(c[v], s0[v], bi0[v]), 0.0f);

        // ---- re-layout hidden C/D -> B operands: half-wave swap via
        //      ds_swizzle_b32 SWAPX16 (single DS op, no index setup)
        float hx[8];
#pragma unroll
        for (int v = 0; v < 8; ++v)
            hx[v] = half_swap(h[v]);

        v2f bb[4];
        // k0=0 : rows 0,1 (lo own) / rows 2,3 (hi: from lane-16)
        bb[0].x = hi ? hx[2] : h[0];  bb[0].y = hi ? hx[3] : h[1];
        // k0=4 : rows 4,5 (lo own) / rows 6,7 (hi: from lane-16)
        bb[1].x = hi ? hx[6] : h[4];  bb[1].y = hi ? hx[7] : h[5];
        // k0=8 : rows 8,9 (lo: from lane+16) / rows 10,11 (hi own)
        bb[2].x = hi ? h[2] : hx[0];  bb[2].y = hi ? h[3] : hx[1];
        // k0=12: rows 12,13 (lo: from lane+16) / rows 14,15 (hi own)
        bb[3].x = hi ? h[6] : hx[4];  bb[3].y = hi ? h[7] : hx[5];

        // ---- layer 2 GEMM + BN + ReLU
        v8f d = {};
#pragma unroll
        for (int j = 0; j < 4; ++j)
            d = __builtin_amdgcn_wmma_f32_16x16x4_f32(false, aw[j], false, bb[j],
                                                      (short)0, d, false, false);

        // ---- layer 3 on lanes 0..15 only (rows 0..7 of C/D live there)
        float g[8];
#pragma unroll
        for (int v = 0; v < 8; ++v)
            g[v] = fmaxf(fmaf(d[v], s1[v], bi1[v]), 0.0f);

        float o = bs;
#pragma unroll
        for (int v = 0; v < 8; ++v)
            o = fmaf(ws[v], g[v], o);

        if (!hi)
            sim[p] = o;   // sim flat index == point index (b*DHW + rem)
    }
}

// ---------------------------------------------------------------------------
// Kernel 2: 9-neighbor dual-dilation aggregation with reflect (even) padding.
// out[b,d,y,x] = weight[b,y,x] * sum_s 0.5*( offset[b,s+9]*sim[y±2,x±2]
//                                          + offset[b,s  ]*sim[y±4,x±4] )
// ---------------------------------------------------------------------------
__device__ __forceinline__ int reflect_ix(int t, int n) {
    t = (t < 0) ? -t : t;
    t = (t >= n) ? (2 * n - 2 - t) : t;
    return t;
}

__global__ __launch_bounds__(256) void aggregate_kernel(
    const float* __restrict__ sim,
    const float* __restrict__ offset,
    const float* __restrict__ weight,
    float* __restrict__ out)
{
    const int idx = blockIdx.x * 256 + threadIdx.x;      // [0, NPTS)
    const int x = idx % NW;
    int t = idx / NW;
    const int y = t % NH;
    t /= NH;
    const int d = t % ND;
    const int b = t / ND;

    int yn[3], xn[3], yw[3], xw[3];
#pragma unroll
    for (int i = 0; i < 3; ++i) {
        yn[i] = reflect_ix(y + (i - 1) * 2, NH);
        xn[i] = reflect_ix(x + (i - 1) * 2, NW);
        yw[i] = reflect_ix(y + (i - 1) * 4, NH);
        xw[i] = reflect_ix(x + (i - 1) * 4, NW);
    }

    const float* sb = sim + (size_t)(b * ND + d) * NH * NW;
    const float* ob = offset + (size_t)b * 18 * NH * NW + y * NW + x;

    float acc = 0.0f;
#pragma unroll
    for (int s = 0; s < 9; ++s) {
        const int iy = s / 3, ix = s % 3;
        const float c1 = sb[yn[iy] * NW + xn[ix]];               // narrow (±2)
        const float c2 = sb[yw[iy] * NW + xw[ix]];               // wide   (±4)
        const float w_wide   = ob[(size_t)s * NH * NW];
        const float w_narrow = ob[(size_t)(s + 9) * NH * NW];
        acc = fmaf(0.5f, fmaf(w_narrow, c1, w_wide * c2), acc);
    }
    out[idx] = acc * weight[(size_t)b * NH * NW + y * NW + x];
}

// ---------------------------------------------------------------------------
// Host-side launch. Input order (setup_inputs): x1, offset, weight, w0,
// bn0_scale, bn0_bias, w1, bn1_scale, bn1_bias, w_sim, b_sim. All float32.
// ---------------------------------------------------------------------------
extern "C" void kernel_launch(void* const* d_in, const int* in_sizes, int n_in,
                              void* d_out, int out_size, void* d_ws, size_t ws_size,
                              hipStream_t stream)
{
    const float* x1        = (const float*)d_in[0];
    const float* offset    = (const float*)d_in[1];
    const float* weight    = (const float*)d_in[2];
    const float* w0        = (const float*)d_in[3];
    const float* bn0_scale = (const float*)d_in[4];
    const float* bn0_bias  = (const float*)d_in[5];
    const float* w1        = (const float*)d_in[6];
    const float* bn1_scale = (const float*)d_in[7];
    const float* bn1_bias  = (const float*)d_in[8];
    const float* w_sim     = (const float*)d_in[9];
    const float* b_sim     = (const float*)d_in[10];
    float* out = (float*)d_out;
    float* sim = (float*)d_ws;      // NPTS * 4 bytes = 5.24 MB scratch

    // Kernel 1: NPTS points, 16 per tile, TILES_PER_WAVE per wave, 8 waves/block
    const int waves   = NPTS / (16 * TILES_PER_WAVE);  // 20480
    const int blocks1 = waves / 8;                     // 2560
    mlp_sim_kernel<<<blocks1, 256, 0, stream>>>(
        x1, w0, bn0_scale, bn0_bias, w1, bn1_scale, bn1_bias, w_sim, b_sim, sim);

    // Kernel 2: one thread per output element
    const int blocks2 = NPTS / 256;                    // 5120
    aggregate_kernel<<<blocks2, 256, 0, stream>>>(sim, offset, weight, out);
}